// SemanticDecoupling_67216238183192
// MI455X (gfx1250) — compile-verified
//
#include <hip/hip_runtime.h>
#include <math.h>

// ---------------------------------------------------------------------------
// SemanticDecoupling for MI455X (gfx1250, wave32, WMMA).
//
// Algebraic simplification: fc3 -> fc4-dot is linear, so
//   coef[m,n] = sum_d v[d] * tanh(img_p[m,d] * word_p[n,d]) + c0,
//   v[d] = sum_e fc4_w[e]*fc3_w[e,d],  c0 constant -> cancels in softmax.
// This removes the 132-GFLOP fc3 GEMM and the 257MB intermediate tensor.
// Remaining GEMMs run on V_WMMA_F32_16X16X4_F32 (fp32-exact matrix path).
// ---------------------------------------------------------------------------

#define BB 4
#define CC 2048
#define HW 196          // 14*14
#define MM (BB*HW)      // 784 = 49*16
#define NCLS 80         // 5*16
#define WD 300          // 75*4
#define ID 1024         // 64*16

typedef __attribute__((ext_vector_type(2))) float v2f;
typedef __attribute__((ext_vector_type(8))) float v8f;

__device__ __forceinline__ float fast_tanh(float x) {
    // tanh(x) = 1 - 2/(exp2(2*log2e*x)+1); saturates correctly for |x| large.
    float e = __builtin_amdgcn_exp2f(x * 2.8853900817779268f);
    return 1.0f - 2.0f * __builtin_amdgcn_rcpf(e + 1.0f);
}

// ---------------- v[d] = sum_e fc4_w[e] * fc3_w[e,d] -----------------------
__global__ void k_vvec(const float* __restrict__ fc3_w,
                       const float* __restrict__ fc4_w,
                       float* __restrict__ v) {
    int d = blockIdx.x * 256 + threadIdx.x;   // 4 blocks * 256 = 1024
    float acc = 0.0f;
    for (int e = 0; e < ID; ++e)
        acc = fmaf(fc4_w[e], fc3_w[(size_t)e * ID + d], acc);
    v[d] = acc;
}

// ---------------- word_p[80,1024] = word @ fc2_w^T (WMMA f32) --------------
// grid (16, 5), block 128 (4 waves; wave w owns n-tile blockIdx.x*4+w)
__global__ void k_wordp(const float* __restrict__ word,   // [80,300]
                        const float* __restrict__ fc2_w,  // [1024,300]
                        float* __restrict__ word_p) {     // [80,1024]
    int lane = threadIdx.x & 31;
    int wave = threadIdx.x >> 5;
    int half = lane >> 4, lm = lane & 15;
    int n0 = (blockIdx.x * 4 + wave) * 16;   // over INT_D
    int m0 = blockIdx.y * 16;                // over N_CLS
    const float* Arow = word  + (size_t)(m0 + lm) * WD;   // A[m,k]
    const float* Brow = fc2_w + (size_t)(n0 + lm) * WD;   // B[k,n]=fc2_w[n,k]
    v8f c = {};
    for (int k = 0; k < WD; k += 4) {
        int kk = k + half * 2;
        v2f a = *(const v2f*)(Arow + kk);
        v2f b = *(const v2f*)(Brow + kk);
        c = __builtin_amdgcn_wmma_f32_16x16x4_f32(false, a, false, b,
                                                  (short)0, c, false, false);
    }
    for (int r = 0; r < 8; ++r)
        word_p[(size_t)(m0 + r + half * 8) * ID + n0 + lm] = c[r];
}

// ---------------- img_p[784,1024] = x @ fc1_w^T (WMMA f32) -----------------
// A[m,k] = img[b, c=k, hw]  (m = b*196+hw), B[k,n] = fc1_w[n,k]
// grid (16, 49), block 128
__global__ void k_fc1(const float* __restrict__ img,     // [4,2048,196]
                      const float* __restrict__ fc1_w,   // [1024,2048]
                      float* __restrict__ img_p) {       // [784,1024]
    int lane = threadIdx.x & 31;
    int wave = threadIdx.x >> 5;
    int half = lane >> 4, lm = lane & 15;
    int n0 = (blockIdx.x * 4 + wave) * 16;   // over INT_D
    int m0 = blockIdx.y * 16;                // over B*HW
    int m  = m0 + lm;
    int b  = m / HW, hw = m - b * HW;
    const float* Abase = img + (size_t)b * CC * HW + hw;  // +k*HW
    const float* Brow  = fc1_w + (size_t)(n0 + lm) * CC;  // +k
    v8f c = {};
    for (int k = 0; k < CC; k += 4) {
        int kk = k + half * 2;
        v2f a;
        a.x = Abase[(size_t)kk * HW];
        a.y = Abase[(size_t)(kk + 1) * HW];
        v2f bfr = *(const v2f*)(Brow + kk);
        c = __builtin_amdgcn_wmma_f32_16x16x4_f32(false, a, false, bfr,
                                                  (short)0, c, false, false);
    }
    for (int r = 0; r < 8; ++r)
        img_p[(size_t)(m0 + r + half * 8) * ID + n0 + lm] = c[r];
}

// ------------- coef[b,n,hw] = sum_d v[d]*tanh(img_p[m,d]*word_p[n,d]) ------
// grid (784, 10), block 256 = 8 waves; wave w handles class n = by*8+w
__global__ void k_coef(const float* __restrict__ img_p,
                       const float* __restrict__ word_p,
                       const float* __restrict__ v,
                       float* __restrict__ coef) {       // [4,80,196]
    __shared__ float sImg[ID];
    __shared__ float sV[ID];
    int m = blockIdx.x;
    for (int i = threadIdx.x; i < ID; i += 256) {
        sImg[i] = img_p[(size_t)m * ID + i];
        sV[i]   = v[i];
    }
    __syncthreads();
    int lane = threadIdx.x & 31;
    int n = blockIdx.y * 8 + (threadIdx.x >> 5);
    const float* wp = word_p + (size_t)n * ID;
    float acc = 0.0f;
    for (int d = lane; d < ID; d += 32)
        acc = fmaf(sV[d], fast_tanh(sImg[d] * wp[d]), acc);
    for (int off = 16; off > 0; off >>= 1)
        acc += __shfl_xor(acc, off, 32);
    if (lane == 0) {
        int b = m / HW, hw = m - b * HW;
        coef[((size_t)b * NCLS + n) * HW + hw] = acc;
    }
}

// ---------------- softmax over 196 spatial positions per (b,n) -------------
__global__ void k_softmax(float* __restrict__ coef) {    // in-place, 320 blocks
    __shared__ float red[256];
    int t = threadIdx.x;
    float* slice = coef + (size_t)blockIdx.x * HW;
    float val = (t < HW) ? slice[t] : -__builtin_inff();
    red[t] = val; __syncthreads();
    for (int s = 128; s > 0; s >>= 1) {
        if (t < s) red[t] = fmaxf(red[t], red[t + s]);
        __syncthreads();
    }
    float mx = red[0]; __syncthreads();
    float e = (t < HW) ? __builtin_amdgcn_exp2f((val - mx) * 1.4426950408889634f) : 0.0f;
    red[t] = e; __syncthreads();
    for (int s = 128; s > 0; s >>= 1) {
        if (t < s) red[t] += red[t + s];
        __syncthreads();
    }
    float rs = __builtin_amdgcn_rcpf(red[0]);
    if (t < HW) slice[t] = e * rs;
}

// -------- out[b,n,c] = sum_hw coef[b,n,hw]*img[b,c,hw]  (WMMA f32) ---------
// A[m=n_cls,k=hw] = coef[b,n,hw] ; B[k=hw,n=c] = img[b,c,hw]; K=196=49*4
// grid (32, 5, 4), block 128
__global__ void k_out(const float* __restrict__ coef,    // [4,80,196]
                      const float* __restrict__ img,     // [4,2048,196]
                      float* __restrict__ out) {         // [4,80,2048]
    int lane = threadIdx.x & 31;
    int wave = threadIdx.x >> 5;
    int half = lane >> 4, lm = lane & 15;
    int n0 = (blockIdx.x * 4 + wave) * 16;   // over C
    int m0 = blockIdx.y * 16;                // over N_CLS
    int b  = blockIdx.z;
    const float* Arow = coef + ((size_t)b * NCLS + m0 + lm) * HW;  // +k
    const float* Brow = img  + ((size_t)b * CC + n0 + lm) * HW;    // +k
    v8f c = {};
    for (int k = 0; k < HW; k += 4) {
        int kk = k + half * 2;
        v2f a   = *(const v2f*)(Arow + kk);
        v2f bfr = *(const v2f*)(Brow + kk);
        c = __builtin_amdgcn_wmma_f32_16x16x4_f32(false, a, false, bfr,
                                                  (short)0, c, false, false);
    }
    for (int r = 0; r < 8; ++r)
        out[((size_t)b * NCLS + m0 + r + half * 8) * CC + n0 + lm] = c[r];
}

extern "C" void kernel_launch(void* const* d_in, const int* in_sizes, int n_in,
                              void* d_out, int out_size, void* d_ws, size_t ws_size,
                              hipStream_t stream) {
    const float* img    = (const float*)d_in[0];  // [4,2048,14,14]
    const float* word   = (const float*)d_in[1];  // [80,300]
    const float* fc1_w  = (const float*)d_in[2];  // [1024,2048]
    const float* fc2_w  = (const float*)d_in[3];  // [1024,300]
    const float* fc3_w  = (const float*)d_in[4];  // [1024,1024]
    // fc3_b (d_in[5]), fc4_b (d_in[7]) cancel in the spatial softmax.
    const float* fc4_w  = (const float*)d_in[6];  // [1,1024]
    float* out = (float*)d_out;                   // [4,80,2048]

    float* ws     = (float*)d_ws;
    float* img_p  = ws;                        // 784*1024
    float* word_p = img_p  + (size_t)MM * ID;  // 80*1024
    float* vvec   = word_p + (size_t)NCLS * ID;// 1024
    float* coef   = vvec + ID;                 // 4*80*196

    k_vvec   <<<dim3(4),          dim3(256), 0, stream>>>(fc3_w, fc4_w, vvec);
    k_wordp  <<<dim3(16, 5),      dim3(128), 0, stream>>>(word, fc2_w, word_p);
    k_fc1    <<<dim3(16, 49),     dim3(128), 0, stream>>>(img, fc1_w, img_p);
    k_coef   <<<dim3(MM, 10),     dim3(256), 0, stream>>>(img_p, word_p, vvec, coef);
    k_softmax<<<dim3(BB * NCLS),  dim3(256), 0, stream>>>(coef);
    k_out    <<<dim3(32, 5, BB),  dim3(128), 0, stream>>>(coef, img, out);
}